// SkipGram_47828755808429
// MI455X (gfx1250) — compile-verified
//
#include <hip/hip_runtime.h>
#include <hip/hip_bf16.h>

typedef __attribute__((ext_vector_type(2))) float v2f;
typedef __attribute__((ext_vector_type(8))) float v8f;

#define VV 100000
#define DD 512
#define BB 1024
#define CC 20
#define NN 64
#define NROWS (CC + NN) /* 84 */

// ---------------------------------------------------------------------------
// Kernel 1: inclusive prefix sum of sample_prob -> cdf (single block, chunked)
// ---------------------------------------------------------------------------
__global__ void __launch_bounds__(1024) sg_scan_kernel(const float* __restrict__ p,
                                                       float* __restrict__ cdf) {
    __shared__ float buf[1024];
    __shared__ float carry_s;
    const int tid = threadIdx.x;
    if (tid == 0) carry_s = 0.0f;
    __syncthreads();
    for (int base = 0; base < VV; base += 1024) {
        float cl = carry_s;
        int i = base + tid;
        float x = (i < VV) ? p[i] : 0.0f;
        buf[tid] = x;
        __syncthreads();
        for (int off = 1; off < 1024; off <<= 1) {
            float t = (tid >= off) ? buf[tid - off] : 0.0f;
            __syncthreads();
            buf[tid] += t;
            __syncthreads();
        }
        float incl = buf[tid] + cl;
        if (i < VV) cdf[i] = incl;
        __syncthreads();
        if (tid == 1023) carry_s = incl;
        __syncthreads();
    }
}

// ---------------------------------------------------------------------------
// Kernel 2: weighted negative sampling via inverse-CDF (deterministic hash RNG)
// ---------------------------------------------------------------------------
__device__ inline unsigned int sg_mix(unsigned long long z) {
    z += 0x9e3779b97f4a7c15ull;
    z = (z ^ (z >> 30)) * 0xbf58476d1ce4e5b9ull;
    z = (z ^ (z >> 27)) * 0x94d049bb133111ebull;
    z = z ^ (z >> 31);
    return (unsigned int)(z >> 32);
}

__global__ void __launch_bounds__(256) sg_sample_kernel(const float* __restrict__ cdf,
                                                        int* __restrict__ neg_idx) {
    int gid = blockIdx.x * blockDim.x + threadIdx.x;
    if (gid >= BB * NN) return;
    float total = cdf[VV - 1];
    unsigned int h = sg_mix(0x2a00000000ull ^ ((unsigned long long)gid * 0x632be59bd9b4e019ull));
    float u = (float)(h >> 8) * (1.0f / 16777216.0f);   // [0,1)
    float target = u * total;
    int lo = 0, hi = VV - 1;
    while (lo < hi) {
        int mid = (lo + hi) >> 1;
        if (cdf[mid] < target) lo = mid + 1; else hi = mid;
    }
    neg_idx[gid] = lo;
}

// ---------------------------------------------------------------------------
// Kernel 3: batched GEMV via V_WMMA_F32_16X16X4_F32.
//   Block = 1 batch row, 64 threads (2 waves), wave w owns tiles 3w..3w+2.
//   A tile row m = gathered embedding row (16 rows/tile), B = center vector
//   replicated across all 16 columns -> every output column is the dot product.
// ---------------------------------------------------------------------------
__device__ inline float sel8(v8f c, int li) {
    float v = c[0];
    v = (li == 1) ? c[1] : v;
    v = (li == 2) ? c[2] : v;
    v = (li == 3) ? c[3] : v;
    v = (li == 4) ? c[4] : v;
    v = (li == 5) ? c[5] : v;
    v = (li == 6) ? c[6] : v;
    v = (li == 7) ? c[7] : v;
    return v;
}

__global__ void __launch_bounds__(64) sg_dot_kernel(const int* __restrict__ center_ids,
                                                    const int* __restrict__ context_ids,
                                                    const float* __restrict__ W_center,
                                                    const float* __restrict__ W_context,
                                                    const int* __restrict__ neg_idx,
                                                    float* __restrict__ out) {
    __shared__ float sC[DD];
    const int b    = blockIdx.x;
    const int tid  = threadIdx.x;
    const int lane = tid & 31;
    const int wave = tid >> 5;
    const int li   = lane & 15;
    const int half = lane >> 4;

    // Stage center vector (2 KB) in LDS once per block.
    const float* crow = W_center + (size_t)center_ids[b] * DD;
    for (int i = tid; i < DD; i += 64) sC[i] = crow[i];
    __syncthreads();

    // Per-lane row pointers for this wave's 3 tiles (lane li -> tile row li).
    const float* p0; const float* p1; const float* p2;
    {
        const float* rp[3];
        for (int t = 0; t < 3; ++t) {
            int j  = (wave * 3 + t) * 16 + li;            // 0..95
            int sj = (j < NROWS) ? j : (NROWS - 1);       // pad rows re-read last row
            const float* table; int id;
            if (sj < CC) { id = context_ids[b * CC + sj]; table = W_center; } // faithful quirk
            else         { id = neg_idx[b * NN + (sj - CC)]; table = W_context; }
            rp[t] = table + (size_t)id * DD + 2 * half;   // fold K half-offset into base
        }
        p0 = rp[0]; p1 = rp[1]; p2 = rp[2];
    }

    // K loop: 128 steps of K=4, three independent WMMA accumulator chains.
    v8f acc0 = {}, acc1 = {}, acc2 = {};
    #pragma unroll 4
    for (int k0 = 0; k0 < DD; k0 += 4) {
        v2f bv = *(const v2f*)&sC[k0 + 2 * half];         // B: center[k] on all columns
        v2f a0 = *(const v2f*)(p0 + k0);
        v2f a1 = *(const v2f*)(p1 + k0);
        v2f a2 = *(const v2f*)(p2 + k0);
        acc0 = __builtin_amdgcn_wmma_f32_16x16x4_f32(false, a0, false, bv, (short)0, acc0, false, false);
        acc1 = __builtin_amdgcn_wmma_f32_16x16x4_f32(false, a1, false, bv, (short)0, acc1, false, false);
        acc2 = __builtin_amdgcn_wmma_f32_16x16x4_f32(false, a2, false, bv, (short)0, acc2, false, false);
    }

    // Emit: C/D layout -> VGPR r holds (M=r, lanes 0-15) and (M=r+8, lanes 16-31);
    // all columns identical, so lane (half*16 + li), li<8 owns output row half*8+li.
    float* out_pos = out;                    // [B, C]
    float* out_neg = out + (size_t)BB * CC;  // [B, NEG]
    if (li < 8) {
        #pragma unroll
        for (int t = 0; t < 3; ++t) {
            v8f c = (t == 0) ? acc0 : (t == 1) ? acc1 : acc2;
            int j = (wave * 3 + t) * 16 + half * 8 + li;
            if (j < NROWS) {
                float v = sel8(c, li);
                if (j < CC) out_pos[(size_t)b * CC + j] = v;
                else        out_neg[(size_t)b * NN + (j - CC)] = v;
            }
        }
    }
}

// ---------------------------------------------------------------------------
extern "C" void kernel_launch(void* const* d_in, const int* in_sizes, int n_in,
                              void* d_out, int out_size, void* d_ws, size_t ws_size,
                              hipStream_t stream) {
    const int*   center_ids  = (const int*)d_in[0];
    const int*   context_ids = (const int*)d_in[1];
    const float* W_center    = (const float*)d_in[2];
    const float* W_context   = (const float*)d_in[3];
    const float* sample_prob = (const float*)d_in[4];
    (void)in_sizes; (void)n_in; (void)out_size; (void)ws_size;

    float* cdf     = (float*)d_ws;                                   // V floats
    int*   neg_idx = (int*)((char*)d_ws + (size_t)VV * sizeof(float)); // B*NEG ints

    sg_scan_kernel<<<1, 1024, 0, stream>>>(sample_prob, cdf);
    sg_sample_kernel<<<(BB * NN + 255) / 256, 256, 0, stream>>>(cdf, neg_idx);
    sg_dot_kernel<<<BB, 64, 0, stream>>>(center_ids, context_ids, W_center, W_context,
                                         neg_idx, (float*)d_out);
}